// RandomEQ_5669356830998
// MI455X (gfx1250) — compile-verified
//
#include <hip/hip_runtime.h>
#include <hip/hip_bf16.h>
#include <math.h>

typedef __attribute__((ext_vector_type(2))) float v2f;
typedef __attribute__((ext_vector_type(8))) float v8f;

#define T_SAMP 441000
#define NROWS  64
#define CHUNK  1024
#define NCH    ((T_SAMP + CHUNK - 1) / CHUNK)   // 431 chunks per row
#define FGRP   (NCH / 16)                       // 26 fully-in-range groups of 16 chunks
#define EDGE_START (FGRP * 16)                  // chunk 416: edge region start
#define EDGE_CH    (NCH - EDGE_START)           // 15 edge chunks per row
#define NBLK   (CHUNK / 16)                     // 64 16-sample blocks per chunk

// workspace float offsets
#define WS_COEF 0                                // b0,b1,b2,a1,a2
#define WS_H    8                                // h[0..15]   impulse response
#define WS_H1   24                               // h1[0..15]  response to y[-1]=1
#define WS_H2   40                               // h2[0..15]  response to y[-2]=1
#define WS_PHI  56                               // h1[C-1],h2[C-1],h1[C-2],h2[C-2]
#define WS_PEND 64                               // [64][431][2] zero-state chunk end pairs
#define WS_SIN  (WS_PEND + NROWS * NCH * 2)      // [64][431][2] stitched initial states

__device__ __forceinline__ float clamp1(float v) {
    return fminf(fmaxf(v, -1.0f), 1.0f);
}

// ---------------------------------------------------------------- coeff setup
__global__ void eq_coeff_kernel(const float* freq, const float* gain,
                                const float* Qp, float* ws) {
    if (blockIdx.x != 0 || threadIdx.x != 0) return;
    float f = freq[0], g = gain[0], q = Qp[0];
    float w0    = 2.0f * 3.14159265358979323846f * f / 44100.0f;
    float sw0   = sinf(w0), cw0 = cosf(w0);
    float alpha = sw0 / (2.0f * q);
    float A     = powf(10.0f, g / 40.0f);
    float a0    = 1.0f + alpha / A;
    float b0 = (1.0f + alpha * A) / a0;
    float b1 = (-2.0f * cw0) / a0;
    float b2 = (1.0f - alpha * A) / a0;
    float a1 = (-2.0f * cw0) / a0;
    float a2 = (1.0f - alpha / A) / a0;
    ws[WS_COEF + 0] = b0; ws[WS_COEF + 1] = b1; ws[WS_COEF + 2] = b2;
    ws[WS_COEF + 3] = a1; ws[WS_COEF + 4] = a2;

    // impulse response h[n] of 1/(1 + a1 z^-1 + a2 z^-2); h1[n] = -a1*h[n]-a2*h[n-1],
    // h2[n] = -a2*h[n].
    float hm1 = 0.0f, hc = 1.0f;                 // h[n-1], h[n]
    for (int n = 0; n < CHUNK; ++n) {
        float h1n = -a1 * hc - a2 * hm1;
        float h2n = -a2 * hc;
        if (n < 16) {
            ws[WS_H  + n] = hc;
            ws[WS_H1 + n] = h1n;
            ws[WS_H2 + n] = h2n;
        }
        if (n == CHUNK - 1) { ws[WS_PHI + 0] = h1n; ws[WS_PHI + 1] = h2n; }
        if (n == CHUNK - 2) { ws[WS_PHI + 2] = h1n; ws[WS_PHI + 3] = h2n; }
        float hn = -a1 * hc - a2 * hm1;
        hm1 = hc; hc = hn;
    }
}

// ------------------------------------------------- pass 1: zero-state chunks
__global__ void eq_pass1_kernel(const float* __restrict__ x, float* ws) {
    int tid = blockIdx.x * blockDim.x + threadIdx.x;
    if (tid >= NROWS * NCH) return;
    int r = tid / NCH, c = tid % NCH;
    float b0 = ws[WS_COEF + 0], b1 = ws[WS_COEF + 1], b2 = ws[WS_COEF + 2];
    float a1 = ws[WS_COEF + 3], a2 = ws[WS_COEF + 4];
    const float* rx = x + (size_t)r * T_SAMP;
    int base = c * CHUNK;
    float xm1 = (base - 1 >= 0) ? rx[base - 1] : 0.0f;
    float xm2 = (base - 2 >= 0) ? rx[base - 2] : 0.0f;
    float p1 = 0.0f, p2 = 0.0f;
    int end = base + CHUNK; if (end > T_SAMP) end = T_SAMP;
    // chunk bases and tail are multiples of 4 -> vectorize by float4
    for (int n = base; n < end; n += 4) {
        float4 xv = *(const float4*)(rx + n);
        float u, p;
        u = b0 * xv.x + b1 * xm1 + b2 * xm2;
        p = u - a1 * p1 - a2 * p2;  p2 = p1; p1 = p;
        u = b0 * xv.y + b1 * xv.x + b2 * xm1;
        p = u - a1 * p1 - a2 * p2;  p2 = p1; p1 = p;
        u = b0 * xv.z + b1 * xv.y + b2 * xv.x;
        p = u - a1 * p1 - a2 * p2;  p2 = p1; p1 = p;
        u = b0 * xv.w + b1 * xv.z + b2 * xv.y;
        p = u - a1 * p1 - a2 * p2;  p2 = p1; p1 = p;
        xm2 = xv.z; xm1 = xv.w;
    }
    ws[WS_PEND + (r * NCH + c) * 2 + 0] = p1;    // y[C-1] (zero state)
    ws[WS_PEND + (r * NCH + c) * 2 + 1] = p2;    // y[C-2]
}

// --------------------------------------- stitch: 2x2 affine scan over chunks
__global__ void eq_stitch_kernel(float* ws) {
    int r = threadIdx.x;
    if (blockIdx.x != 0 || r >= NROWS) return;
    float p11 = ws[WS_PHI + 0], p12 = ws[WS_PHI + 1];
    float p21 = ws[WS_PHI + 2], p22 = ws[WS_PHI + 3];
    float s1 = 0.0f, s2 = 0.0f;                  // (y[-1], y[-2]) entering chunk
    for (int c = 0; c < NCH; ++c) {
        int idx = (r * NCH + c) * 2;
        ws[WS_SIN + idx + 0] = s1;
        ws[WS_SIN + idx + 1] = s2;
        float pe0 = ws[WS_PEND + idx + 0];
        float pe1 = ws[WS_PEND + idx + 1];
        float n1 = pe0 + p11 * s1 + p12 * s2;    // true y at chunk end
        float n2 = pe1 + p21 * s1 + p22 * s2;
        s1 = n1; s2 = n2;
    }
}

// ------------------------------- pass 2 (fast): WMMA Toeplitz, groups 0..25
__global__ void eq_pass2_wmma_kernel(const float* __restrict__ x,
                                     float* __restrict__ out,
                                     const float* __restrict__ ws) {
    int r  = blockIdx.x / FGRP;                  // clip row
    int g  = blockIdx.x % FGRP;                  // group of 16 full chunks
    int ln = threadIdx.x;                        // 0..31 (one wave32 per block)
    int m  = ln & 15;                            // matrix row (A/D) or column (B/D)
    int hi = ln >> 4;

    float b0 = ws[WS_COEF + 0], b1 = ws[WS_COEF + 1], b2 = ws[WS_COEF + 2];

    // A = [ L16 | h1 | h2 | 0 | 0 ]  (16 x 20), loop invariant.
    // A layout (ISA 7.12.2, 32-bit A 16x4): lane m = row M; tile t VGPR e holds
    // K = 4t + 2*hi + e.
    v2f a[5];
#pragma unroll
    for (int t = 0; t < 5; ++t) {
#pragma unroll
        for (int e = 0; e < 2; ++e) {
            int k = 4 * t + 2 * hi + e;
            float v = 0.0f;
            if (k < 16)       { if (m >= k) v = ws[WS_H  + (m - k)]; }
            else if (k == 16) { v = ws[WS_H1 + m]; }
            else if (k == 17) { v = ws[WS_H2 + m]; }
            a[t][e] = v;
        }
    }

    int  ch      = g * 16 + m;                   // this column's chunk (< 416 < NCH)
    int  colBase = ch * CHUNK;
    const float* rx = x   + (size_t)r * T_SAMP;
    float*       ry = out + (size_t)r * T_SAMP;

    // initial per-chunk state from the stitch pass (only lanes 0..15 feed B)
    float s1 = ws[WS_SIN + (r * NCH + ch) * 2 + 0];
    float s2 = ws[WS_SIN + (r * NCH + ch) * 2 + 1];

    // carried FIR history x[s0-2], x[s0-1]; chunk 0 starts zero-padded
    float xm2 = 0.0f, xm1 = 0.0f;
    if (colBase >= 2) {
        float2 hm = *(const float2*)(rx + colBase - 2);
        xm2 = hm.x; xm1 = hm.y;
    }

    for (int i = 0; i < NBLK; ++i) {
        int s0 = colBase + i * 16;
        const float* p = rx + s0;
        float4 w0 = *(const float4*)(p +  0);
        float4 w1 = *(const float4*)(p +  4);
        float4 w2 = *(const float4*)(p +  8);
        float4 w3 = *(const float4*)(p + 12);
        __builtin_prefetch(p + 16, 0, 1);        // next block's window

        float X[18];                              // X[j+2] = x[s0+j]
        X[0]  = xm2;  X[1]  = xm1;
        X[2]  = w0.x; X[3]  = w0.y; X[4]  = w0.z; X[5]  = w0.w;
        X[6]  = w1.x; X[7]  = w1.y; X[8]  = w1.z; X[9]  = w1.w;
        X[10] = w2.x; X[11] = w2.y; X[12] = w2.z; X[13] = w2.w;
        X[14] = w3.x; X[15] = w3.y; X[16] = w3.z; X[17] = w3.w;
        xm2 = w3.z; xm1 = w3.w;                   // history for next block

        v2f bt[4];
        if (hi == 0) {                            // k = 4t + e  (constant indices)
#pragma unroll
            for (int t = 0; t < 4; ++t) {
                bt[t][0] = b0 * X[4*t + 2] + b1 * X[4*t + 1] + b2 * X[4*t + 0];
                bt[t][1] = b0 * X[4*t + 3] + b1 * X[4*t + 2] + b2 * X[4*t + 1];
            }
        } else {                                  // k = 4t + 2 + e
#pragma unroll
            for (int t = 0; t < 4; ++t) {
                bt[t][0] = b0 * X[4*t + 4] + b1 * X[4*t + 3] + b2 * X[4*t + 2];
                bt[t][1] = b0 * X[4*t + 5] + b1 * X[4*t + 4] + b2 * X[4*t + 3];
            }
        }

        // B tile 4: rows 16 (s1), 17 (s2), 18/19 zero
        v2f b4;
        b4[0] = (hi == 0) ? s1 : 0.0f;
        b4[1] = (hi == 0) ? s2 : 0.0f;

        v8f acc = {0.f, 0.f, 0.f, 0.f, 0.f, 0.f, 0.f, 0.f};
        acc = __builtin_amdgcn_wmma_f32_16x16x4_f32(false, a[0], false, bt[0],
                                                    (short)0, acc, false, false);
        acc = __builtin_amdgcn_wmma_f32_16x16x4_f32(false, a[1], false, bt[1],
                                                    (short)0, acc, false, false);
        acc = __builtin_amdgcn_wmma_f32_16x16x4_f32(false, a[2], false, bt[2],
                                                    (short)0, acc, false, false);
        acc = __builtin_amdgcn_wmma_f32_16x16x4_f32(false, a[3], false, bt[3],
                                                    (short)0, acc, false, false);
        acc = __builtin_amdgcn_wmma_f32_16x16x4_f32(false, a[4], false, b4,
                                                    (short)0, acc, false, false);

        // next-block state: D row 15 (VGPR7 lanes 16..31), row 14 (VGPR6 lanes 16..31)
        s1 = __shfl(acc[7], 16 + m, 32);
        s2 = __shfl(acc[6], 16 + m, 32);

        // lane stores its 8 contiguous outputs: rows 8*hi .. 8*hi+7, 16B aligned
        float4 o0, o1;
        o0.x = clamp1(acc[0]); o0.y = clamp1(acc[1]);
        o0.z = clamp1(acc[2]); o0.w = clamp1(acc[3]);
        o1.x = clamp1(acc[4]); o1.y = clamp1(acc[5]);
        o1.z = clamp1(acc[6]); o1.w = clamp1(acc[7]);
        float* q = ry + s0 + 8 * hi;
        *(float4*)(q + 0) = o0;
        *(float4*)(q + 4) = o1;
    }
}

// ---------------------- pass 2 (edge): chunks 416..430, sequential + stores
__global__ void eq_pass2_edge_kernel(const float* __restrict__ x,
                                     float* __restrict__ out,
                                     const float* __restrict__ ws) {
    int tid = blockIdx.x * blockDim.x + threadIdx.x;
    if (tid >= NROWS * EDGE_CH) return;
    int r = tid / EDGE_CH;
    int c = EDGE_START + tid % EDGE_CH;
    float b0 = ws[WS_COEF + 0], b1 = ws[WS_COEF + 1], b2 = ws[WS_COEF + 2];
    float a1 = ws[WS_COEF + 3], a2 = ws[WS_COEF + 4];
    const float* rx = x   + (size_t)r * T_SAMP;
    float*       ry = out + (size_t)r * T_SAMP;
    int base = c * CHUNK;                        // >= 2 always here
    float y1 = ws[WS_SIN + (r * NCH + c) * 2 + 0];
    float y2 = ws[WS_SIN + (r * NCH + c) * 2 + 1];
    float xm1 = rx[base - 1];
    float xm2 = rx[base - 2];
    int end = base + CHUNK; if (end > T_SAMP) end = T_SAMP;
    for (int n = base; n < end; n += 4) {        // tail (680) is a multiple of 4
        float4 xv = *(const float4*)(rx + n);
        float4 o;
        float u, y;
        u = b0 * xv.x + b1 * xm1 + b2 * xm2;
        y = u - a1 * y1 - a2 * y2;  y2 = y1; y1 = y;  o.x = clamp1(y);
        u = b0 * xv.y + b1 * xv.x + b2 * xm1;
        y = u - a1 * y1 - a2 * y2;  y2 = y1; y1 = y;  o.y = clamp1(y);
        u = b0 * xv.z + b1 * xv.y + b2 * xv.x;
        y = u - a1 * y1 - a2 * y2;  y2 = y1; y1 = y;  o.z = clamp1(y);
        u = b0 * xv.w + b1 * xv.z + b2 * xv.y;
        y = u - a1 * y1 - a2 * y2;  y2 = y1; y1 = y;  o.w = clamp1(y);
        xm2 = xv.z; xm1 = xv.w;
        *(float4*)(ry + n) = o;
    }
}

// ---------------------------------------------------------------------- host
extern "C" void kernel_launch(void* const* d_in, const int* in_sizes, int n_in,
                              void* d_out, int out_size, void* d_ws, size_t ws_size,
                              hipStream_t stream) {
    const float* clip = (const float*)d_in[0];
    const float* freq = (const float*)d_in[1];
    const float* gain = (const float*)d_in[2];
    const float* Q    = (const float*)d_in[3];
    float* out = (float*)d_out;
    float* ws  = (float*)d_ws;
    (void)in_sizes; (void)n_in; (void)out_size; (void)ws_size;

    eq_coeff_kernel<<<1, 1, 0, stream>>>(freq, gain, Q, ws);

    int p1_threads = NROWS * NCH;
    eq_pass1_kernel<<<(p1_threads + 255) / 256, 256, 0, stream>>>(clip, ws);

    eq_stitch_kernel<<<1, NROWS, 0, stream>>>(ws);

    eq_pass2_wmma_kernel<<<NROWS * FGRP, 32, 0, stream>>>(clip, out, ws);

    int edge_threads = NROWS * EDGE_CH;
    eq_pass2_edge_kernel<<<(edge_threads + 255) / 256, 256, 0, stream>>>(clip, out, ws);
}